// DynEmbedding_75265006895642
// MI455X (gfx1250) — compile-verified
//
#include <hip/hip_runtime.h>

// Embedding gather: out[row, :] = weight[x[row], :]
//   x:      [819200] int32 (4096*200), values in [0, 1'000'000)
//   weight: [1'000'000, 128] float32
//   out:    [819200, 128] float32
//
// MI455X design: memory-bound (~840 MB total @ 23.3 TB/s -> ~36 us floor).
// wave32-native tiling: one wave per 4 rows (whole wave on each row); 32 lanes * float4 = 512 B = one row,
// so each row is exactly one global_load_b128 + one global_store_b128 per lane.
//
// Round-1 fixes vs previous version:
//  * row base forced into an SGPR via readfirstlane -> scalar bounds check
//    (no per-row exec-mask branches), scalar index addresses (single
//    s_load_b128 for 4 indices instead of 4 serialized global_load_b32 +
//    s_wait_loadcnt pairs), and SADDR-form B128 row loads/stores.
//  * wave-granular fast path is completely branch-free.

typedef float __attribute__((ext_vector_type(4))) f32x4;
typedef int   __attribute__((ext_vector_type(4))) i32x4;

#define EMB_DIM 128
#define ROWS_PER_WAVE 4
#define BLOCK_THREADS 256               // 8 waves per block (wave32)
#define ROWS_PER_BLOCK ((BLOCK_THREADS / 32) * ROWS_PER_WAVE)  // 32

__global__ __launch_bounds__(BLOCK_THREADS) void
dyn_embedding_gather_kernel(const int* __restrict__ x,
                            const float* __restrict__ weight,
                            float* __restrict__ out,
                            int nrows)
{
    const int lane = threadIdx.x & 31;               // lane within wave32

    // Wave-uniform row base, forced into an SGPR so everything derived from
    // it (bounds check, index load address, output base) is scalar.
    const int waveId = (int)(blockIdx.x * (BLOCK_THREADS >> 5)) + (int)(threadIdx.x >> 5);
    const int row0   = __builtin_amdgcn_readfirstlane(waveId * ROWS_PER_WAVE);

    if (row0 + ROWS_PER_WAVE <= nrows) {
        // ---- fast path: fully branch-free -------------------------------
        // 4 consecutive wave-uniform indices: scalar address -> s_load_b128.
        const i32x4 iv = *(const i32x4*)(x + row0);

        // Cluster the 4 row loads (B128, SADDR + lane*16) for 4-deep MLP.
        f32x4 v[ROWS_PER_WAVE];
#pragma unroll
        for (int r = 0; r < ROWS_PER_WAVE; ++r) {
            const int idx = __builtin_amdgcn_readfirstlane(iv[r]);
            const f32x4* src = (const f32x4*)(weight + (long long)idx * EMB_DIM);
            v[r] = src[lane];                        // global_load_b128 (RT)
        }

        // Streamed non-temporal stores (B128, th:TH_STORE_NT).
        f32x4* dst0 = (f32x4*)(out + (long long)row0 * EMB_DIM);
#pragma unroll
        for (int r = 0; r < ROWS_PER_WAVE; ++r) {
            __builtin_nontemporal_store(v[r], dst0 + r * (EMB_DIM / 4) + lane);
        }
    } else {
        // ---- tail path (never taken for 819200 rows; kept for safety) ---
#pragma unroll
        for (int r = 0; r < ROWS_PER_WAVE; ++r) {
            const int row = row0 + r;                // scalar compare/branch
            if (row < nrows) {
                const int idx = __builtin_amdgcn_readfirstlane(x[row]);
                const f32x4* src = (const f32x4*)(weight + (long long)idx * EMB_DIM);
                f32x4 val = src[lane];
                f32x4* dst = (f32x4*)(out + (long long)row * EMB_DIM);
                __builtin_nontemporal_store(val, dst + lane);
            }
        }
    }
}

extern "C" void kernel_launch(void* const* d_in, const int* in_sizes, int n_in,
                              void* d_out, int out_size, void* d_ws, size_t ws_size,
                              hipStream_t stream) {
    (void)n_in; (void)d_ws; (void)ws_size; (void)out_size;

    const int*   x      = (const int*)d_in[0];     // [BATCH*HIST] int32
    const float* weight = (const float*)d_in[1];   // [VOCAB, 128] float32
    float*       out    = (float*)d_out;           // [BATCH*HIST, 128] float32

    const int nrows = in_sizes[0];                 // 4096*200 = 819200

    const int blocks = (nrows + ROWS_PER_BLOCK - 1) / ROWS_PER_BLOCK;  // 25600
    dyn_embedding_gather_kernel<<<blocks, BLOCK_THREADS, 0, stream>>>(
        x, weight, out, nrows);
}